// Fusion_block_32315333935155
// MI455X (gfx1250) — compile-verified
//
#include <hip/hip_runtime.h>
#include <math.h>

#define ALPHA  0.85f
#define THRESH 0.6f
#define B_     32
#define N_     1024
#define D_     256
#define NTOL   (N_ * 1e-3f)
#define MAX_ITER 100

typedef float v2f __attribute__((ext_vector_type(2)));
typedef float v8f __attribute__((ext_vector_type(8)));

#ifndef __has_builtin
#define __has_builtin(x) 0
#endif
#if __has_builtin(__builtin_amdgcn_wmma_f32_16x16x4_f32)
#define USE_WMMA 1
#else
#define USE_WMMA 0
#endif

// ---------------- Phase 1: per-row stats: nstart = sum_d x, deg = sum_m A ----
__global__ void k_row_stats(const float* __restrict__ x, const float* __restrict__ adj,
                            float* __restrict__ nstart, float* __restrict__ deg) {
    const int n = blockIdx.x, b = blockIdx.y, t = threadIdx.x;
    __shared__ float r0[256], r1[256];
    float xs = x[(((size_t)b * N_) + n) * D_ + t];           // D_==256, one per thread
    float ds = 0.f;
    const float* arow = adj + (((size_t)b * N_) + n) * (size_t)N_;
    for (int j = t; j < N_; j += 256) {
        float a = arow[j];
        ds += (a >= THRESH) ? a : 0.f;
    }
    r0[t] = xs; r1[t] = ds; __syncthreads();
    for (int s = 128; s > 0; s >>= 1) {
        if (t < s) { r0[t] += r0[t + s]; r1[t] += r1[t + s]; }
        __syncthreads();
    }
    if (t == 0) { nstart[b * N_ + n] = r0[0]; deg[b * N_ + n] = r1[0]; }
}

// ---------------- Phase 2: pr0 = nstart / sum(nstart) (in place) ------------
__global__ void k_norm(float* __restrict__ pr) {
    const int b = blockIdx.x, t = threadIdx.x;
    __shared__ float red[256];
    float s = 0.f;
    for (int i = t; i < N_; i += 256) s += pr[b * N_ + i];
    red[t] = s; __syncthreads();
    for (int st = 128; st > 0; st >>= 1) {
        if (t < st) red[t] += red[t + st];
        __syncthreads();
    }
    float tot = red[0];
    for (int i = t; i < N_; i += 256) pr[b * N_ + i] /= tot;
}

__global__ void k_setflag(int* flag, int v) { *flag = v; }

// ---------------- Phase 3a: q = pr/deg (0 if dangling), cterm ---------------
__global__ void k_qdsum(const float* __restrict__ pr, const float* __restrict__ deg,
                        float* __restrict__ q, float* __restrict__ cterm,
                        const int* __restrict__ flag) {
    if (*flag == 0) return;
    const int b = blockIdx.x, t = threadIdx.x;
    __shared__ float red[256];
    float ds = 0.f;
    for (int i = t; i < N_; i += 256) {
        float d = deg[b * N_ + i], p = pr[b * N_ + i];
        bool dang = (d == 0.f);
        q[b * N_ + i] = dang ? 0.f : p / d;
        if (dang) ds += p;
    }
    red[t] = ds; __syncthreads();
    for (int st = 128; st > 0; st >>= 1) {
        if (t < st) red[t] += red[t + st];
        __syncthreads();
    }
    if (t == 0) cterm[b] = (ALPHA * red[0] + (1.f - ALPHA)) * (1.f / N_);
}

// ---------------- Phase 3b: prn = ALPHA * (A @ q) + cterm  (WMMA tiles) -----
// One wave = 16 output rows. A-frag: 16x4 fp32 tile of thresholded adj.
// B-frag: q[k..k+3] replicated over all 16 columns -> every result column
// holds the matvec result; lanes 0/16 extract rows 0..7 / 8..15.
__global__ void k_matvec(const float* __restrict__ adj, const float* __restrict__ pr,
                         float* __restrict__ prn, const float* __restrict__ q,
                         const float* __restrict__ cterm, float* __restrict__ errpart,
                         const int* __restrict__ flag) {
    const int b = blockIdx.y;
    if (*flag == 0) {                    // converged: keep ping-pong consistent
        int base = blockIdx.x * 64 + threadIdx.x;
        if (threadIdx.x < 64) prn[b * N_ + base] = pr[b * N_ + base];
        return;
    }
    __shared__ float qs[N_];
    for (int i = threadIdx.x; i < N_; i += blockDim.x) qs[i] = q[b * N_ + i];
    __syncthreads();

    const int lane  = threadIdx.x & 31;
    const int wave  = threadIdx.x >> 5;
    const int tile  = blockIdx.x * 4 + wave;     // 0..63
    const int m0    = tile * 16;
    const int hv    = lane >> 4;                 // lane half (K split)
    const int r     = lane & 15;                 // row within tile
    const float* lp = adj + (((size_t)b * N_) + m0 + r) * (size_t)N_ + 2 * hv;
    const float* qp = qs + 2 * hv;
    const float  ct = cterm[b];

#if USE_WMMA
    v8f c0 = {}, c1 = {};
    for (int k = 0; k < N_; k += 8) {
        if ((k & 63) == 0) __builtin_prefetch(lp + k + 512, 0, 1);
        float a0 = lp[k],     a1 = lp[k + 1];
        float a2 = lp[k + 4], a3 = lp[k + 5];
        v2f A0 = { (a0 >= THRESH) ? a0 : 0.f, (a1 >= THRESH) ? a1 : 0.f };
        v2f A1 = { (a2 >= THRESH) ? a2 : 0.f, (a3 >= THRESH) ? a3 : 0.f };
        v2f B0 = { qp[k],     qp[k + 1] };
        v2f B1 = { qp[k + 4], qp[k + 5] };
        c0 = __builtin_amdgcn_wmma_f32_16x16x4_f32(false, A0, false, B0, (short)0, c0, false, false);
        c1 = __builtin_amdgcn_wmma_f32_16x16x4_f32(false, A1, false, B1, (short)0, c1, false, false);
    }
    v8f c = c0 + c1;
    float es = 0.f;
    if (r == 0) {                                  // lanes 0 and 16
        #pragma unroll
        for (int v = 0; v < 8; v++) {
            int m = m0 + v + 8 * hv;
            float val = ALPHA * c[v] + ct;
            prn[b * N_ + m] = val;
            es += fabsf(val - pr[b * N_ + m]);
        }
    }
    float other = __shfl_xor(es, 16, 32);
    if (lane == 0) errpart[b * 64 + tile] = es + other;
#else
    float acc = 0.f;
    for (int k = 0; k < N_; k += 4) {
        if ((k & 63) == 0) __builtin_prefetch(lp + k + 512, 0, 1);
        float a0 = lp[k], a1 = lp[k + 1];
        a0 = (a0 >= THRESH) ? a0 : 0.f;
        a1 = (a1 >= THRESH) ? a1 : 0.f;
        acc += a0 * qp[k] + a1 * qp[k + 1];
    }
    float other = __shfl_xor(acc, 16, 32);         // combine K halves
    float e = 0.f;
    if (hv == 0) {
        int m = m0 + r;
        float val = ALPHA * (acc + other) + ct;
        prn[b * N_ + m] = val;
        e = fabsf(val - pr[b * N_ + m]);
    }
    for (int s = 8; s >= 1; s >>= 1) e += __shfl_xor(e, s, 32);
    if (lane == 0) errpart[b * 64 + tile] = e;
#endif
}

// ---------------- Phase 3c: global convergence flag (deterministic) ---------
__global__ void k_flag(const float* __restrict__ errpart, int* flag) {
    if (*flag == 0) return;
    const int b = threadIdx.x;                     // 32 threads, one per batch
    float s = 0.f;
    for (int t = 0; t < 64; t++) s += errpart[b * 64 + t];   // fixed order
    int f = (s >= NTOL) ? 1 : 0;
    int any = __any(f);
    if (b == 0) *flag = any ? 1 : 0;
}

// ---------------- Phase 4: stable descending argsort via rank counting ------
__global__ void k_rank(const float* __restrict__ pr, int* __restrict__ perm) {
    const int b = blockIdx.y, t = threadIdx.x;
    __shared__ float ps[N_];
    for (int i = t; i < N_; i += 256) ps[i] = pr[b * N_ + i];
    __syncthreads();
    int i = blockIdx.x * 256 + t;
    float my = ps[i];
    int rank = 0;
    for (int j = 0; j < N_; j++) {
        float pj = ps[j];
        rank += (pj > my) || (pj == my && j < i);
    }
    perm[b * N_ + rank] = i;
}

// ---------------- Phase 5: gathers ------------------------------------------
__global__ void k_gather_x(const float* __restrict__ x, const int* __restrict__ perm,
                           float* __restrict__ out) {
    const int rrow = blockIdx.x, b = blockIdx.y, t = threadIdx.x;  // D_ threads
    int src = perm[b * N_ + rrow];
    out[(((size_t)b * N_) + rrow) * D_ + t] = x[(((size_t)b * N_) + src) * D_ + t];
}

__global__ void k_gather_adj(const float* __restrict__ adj, const int* __restrict__ perm,
                             float* __restrict__ out) {
    const int rrow = blockIdx.x, b = blockIdx.y, t = threadIdx.x;
    __shared__ int ps[N_];
    for (int i = t; i < N_; i += 256) ps[i] = perm[b * N_ + i];
    __syncthreads();
    int src = ps[rrow];
    const float* row = adj + (((size_t)b * N_) + src) * (size_t)N_;
    float* orow = out + (((size_t)b * N_) + rrow) * (size_t)N_;
    for (int c = t; c < N_; c += 256) orow[c] = row[ps[c]];
}

// ---------------- Orchestration ---------------------------------------------
static void run_pair(const float* x, const float* adj, float* xo, float* ao,
                     float* ws, hipStream_t stream) {
    const int BN = B_ * N_;
    float* prA     = ws;                 // [B,N]
    float* prB     = prA + BN;           // [B,N]
    float* deg     = prB + BN;           // [B,N]
    float* q       = deg + BN;           // [B,N]
    float* cterm   = q + BN;             // [B]
    float* errpart = cterm + B_;         // [B,64]
    int*   flag    = (int*)(errpart + B_ * 64);
    int*   perm    = flag + 4;           // [B,N] ints

    k_row_stats<<<dim3(N_, B_), 256, 0, stream>>>(x, adj, prA, deg);
    k_norm<<<B_, 256, 0, stream>>>(prA);
    k_setflag<<<1, 1, 0, stream>>>(flag, 1);

    float* cur = prA; float* nxt = prB;
    for (int it = 0; it < MAX_ITER; ++it) {
        k_qdsum<<<B_, 256, 0, stream>>>(cur, deg, q, cterm, flag);
        k_matvec<<<dim3(N_ / 64, B_), 128, 0, stream>>>(adj, cur, nxt, q, cterm, errpart, flag);
        k_flag<<<1, 32, 0, stream>>>(errpart, flag);
        float* tmp = cur; cur = nxt; nxt = tmp;
    }

    k_rank<<<dim3(N_ / 256, B_), 256, 0, stream>>>(cur, perm);
    k_gather_x<<<dim3(N_, B_), D_, 0, stream>>>(x, perm, xo);
    k_gather_adj<<<dim3(N_, B_), 256, 0, stream>>>(adj, perm, ao);
}

extern "C" void kernel_launch(void* const* d_in, const int* in_sizes, int n_in,
                              void* d_out, int out_size, void* d_ws, size_t ws_size,
                              hipStream_t stream) {
    const float* x1   = (const float*)d_in[0];
    const float* adj1 = (const float*)d_in[1];
    const float* x3   = (const float*)d_in[2];
    const float* adj3 = (const float*)d_in[3];
    float* out = (float*)d_out;
    const size_t XSZ = (size_t)B_ * N_ * D_;   // 8388608
    const size_t ASZ = (size_t)B_ * N_ * N_;   // 33554432
    float* ws = (float*)d_ws;

    run_pair(x1, adj1, out,              out + XSZ,             ws, stream);
    run_pair(x3, adj3, out + XSZ + ASZ,  out + 2 * XSZ + ASZ,   ws, stream);
}